// Sampler_32865089749571
// MI455X (gfx1250) — compile-verified
//
#include <hip/hip_runtime.h>
#include <hip/hip_bf16.h>
#include <stdint.h>

// Sampler reduces to per-row argmax (temperature/top-p/top-k never mask the
// max token; softmax is monotonic). Streaming argmax over 256 x 128000 f32.
//
// Stage 1: grid (256 rows x 5 segments), 256 threads (8 wave32's).
//   Per wave: 800 float4 = 5 tiles of 5 x global_load_async_to_lds_b128,
//   double-buffered in LDS, synchronized with s_wait_asynccnt.
// Stage 2: one block reduces the 5 partials per row, writes (float)idx.

#define ROWS      256
#define VOCAB_F4  32000          // 128000 floats / 4
#define SEGS      5
#define SEG_F4    (VOCAB_F4 / SEGS)      // 6400
#define WAVES     8
#define WAVE_F4   (SEG_F4 / WAVES)       // 800
#define UNROLL    5                      // b128 loads per lane per tile
#define TILE_F4   (32 * UNROLL)          // 160 float4 per wave-tile
#define ROUNDS    (WAVE_F4 / TILE_F4)    // 5

__device__ __forceinline__ void amax_combine(float& bv, int& bi, float v, int i) {
    if (v > bv || (v == bv && i < bi)) { bv = v; bi = i; }
}

__global__ __launch_bounds__(256) void
sampler_argmax_stage1(const float* __restrict__ logits,
                      float* __restrict__ part_val,
                      int*   __restrict__ part_idx) {
    __shared__ float4 tile[WAVES][2][TILE_F4];   // 40 KB
    __shared__ float  swv[WAVES];
    __shared__ int    swi[WAVES];

    const int row  = blockIdx.x;
    const int seg  = blockIdx.y;
    const int tid  = threadIdx.x;
    const int w    = tid >> 5;
    const int lane = tid & 31;

    // wave-private global base (float4 units within the row)
    const int wave_f4_base = seg * SEG_F4 + w * WAVE_F4;
    const float4* __restrict__ gwave =
        reinterpret_cast<const float4*>(logits + (size_t)row * (VOCAB_F4 * 4))
        + wave_f4_base;

    // LDS byte addresses of this wave's two buffers (low 32 bits of the
    // generic LDS-aperture address == workgroup-relative LDS offset)
    const uint32_t lds_buf0 = (uint32_t)(uintptr_t)(&tile[w][0][0]);
    const uint32_t lds_buf1 = (uint32_t)(uintptr_t)(&tile[w][1][0]);

    // ---- async tile issue: 5 x b128 per lane, one tile = 2560 B / wave ----
    auto issue_tile = [&](int t, uint32_t lds_base) {
        const float4* g = gwave + t * TILE_F4 + lane;
        uint32_t l = lds_base + (uint32_t)(lane * 16);
#pragma unroll
        for (int j = 0; j < UNROLL; ++j) {
            asm volatile("global_load_async_to_lds_b128 %0, %1, off"
                         :: "v"(l), "v"(g) : "memory");
            g += 32;
            l += 32 * 16;
        }
    };

    float bv = -__builtin_inff();
    int   bi = 0x7fffffff;

    issue_tile(0, lds_buf0);

#pragma unroll
    for (int t = 0; t < ROUNDS; ++t) {
        const uint32_t cur = (t & 1) ? lds_buf1 : lds_buf0;
        if (t + 1 < ROUNDS) {
            issue_tile(t + 1, ((t + 1) & 1) ? lds_buf1 : lds_buf0);
            // loads complete in order: <=UNROLL outstanding => tile t landed
            asm volatile("s_wait_asynccnt 0x5" ::: "memory");
        } else {
            asm volatile("s_wait_asynccnt 0x0" ::: "memory");
        }
        const float4* lbuf = (t & 1) ? &tile[w][1][0] : &tile[w][0][0];
        (void)cur;
#pragma unroll
        for (int j = 0; j < UNROLL; ++j) {
            const int f4i = wave_f4_base + t * TILE_F4 + j * 32 + lane;
            float4 v = lbuf[j * 32 + lane];
            const int e = f4i * 4;
            amax_combine(bv, bi, v.x, e + 0);
            amax_combine(bv, bi, v.y, e + 1);
            amax_combine(bv, bi, v.z, e + 2);
            amax_combine(bv, bi, v.w, e + 3);
        }
    }

    // ---- wave32 reduction (ds-permute shuffles) ----
#pragma unroll
    for (int off = 16; off > 0; off >>= 1) {
        float ov = __shfl_xor(bv, off, 32);
        int   oi = __shfl_xor(bi, off, 32);
        amax_combine(bv, bi, ov, oi);
    }
    if (lane == 0) { swv[w] = bv; swi[w] = bi; }
    __syncthreads();

    if (tid == 0) {
        float rv = swv[0]; int ri = swi[0];
#pragma unroll
        for (int i = 1; i < WAVES; ++i) amax_combine(rv, ri, swv[i], swi[i]);
        const int p = row * SEGS + seg;
        part_val[p] = rv;
        part_idx[p] = ri;
    }
}

__global__ __launch_bounds__(256) void
sampler_argmax_stage2(const float* __restrict__ part_val,
                      const int*   __restrict__ part_idx,
                      float* __restrict__ out) {
    const int row = threadIdx.x;   // one block of 256 threads
    float bv = -__builtin_inff();
    int   bi = 0x7fffffff;
#pragma unroll
    for (int s = 0; s < SEGS; ++s) {
        amax_combine(bv, bi, part_val[row * SEGS + s], part_idx[row * SEGS + s]);
    }
    out[row] = (float)bi;          // vocab index, exact in f32 (< 2^24)
}

extern "C" void kernel_launch(void* const* d_in, const int* in_sizes, int n_in,
                              void* d_out, int out_size, void* d_ws, size_t ws_size,
                              hipStream_t stream) {
    (void)in_sizes; (void)n_in; (void)out_size; (void)ws_size;
    const float* logits = (const float*)d_in[0];

    float* part_val = (float*)d_ws;                       // 1280 floats
    int*   part_idx = (int*)(part_val + ROWS * SEGS);     // 1280 ints

    dim3 grid1(ROWS, SEGS);
    sampler_argmax_stage1<<<grid1, 256, 0, stream>>>(logits, part_val, part_idx);
    sampler_argmax_stage2<<<1, 256, 0, stream>>>(part_val, part_idx, (float*)d_out);
}